// lspCalc_84207128805510
// MI455X (gfx1250) — compile-verified
//
#include <hip/hip_runtime.h>
#include <stdint.h>

// ---------------------------------------------------------------------------
// LSP pipeline for MI455X (gfx1250, wave32).
//   out (4,6,1024,1024): [0]=tpiHS [1]=slp [2]=tpiL [3]=hillshade [4]=crvPro [5]=crvPln
//
// Padded row-prefix buffer PP per batch: rows 1154 (64 zero rows above, image
// rows 64..1087, zero rows below), row stride 1152 floats:
//   cols 0..51        : 0              (clamps "x - w < 0" segment starts)
//   cols 52..1076     : Pex[0..1024]   (exclusive prefix, Pex[0]=0)
//   cols 1077..1151   : row total      (clamps "x + w + 1 > 1024" ends)
// Disk sum(y,x) = sum_dy Pex[y+dy][x+w(dy)+1] - Pex[y+dy][x-w(dy)].
// ---------------------------------------------------------------------------

#define W_IMG   1024
#define PP_ROWS 1154
#define PP_PITCH 1152
#define CH_STRIDE (1024*1024)

typedef float v2f  __attribute__((ext_vector_type(2)));
typedef float v8f  __attribute__((ext_vector_type(8)));
typedef unsigned int u32x4 __attribute__((ext_vector_type(4)));
typedef int i32x8 __attribute__((ext_vector_type(8)));
typedef int i32x4 __attribute__((ext_vector_type(4)));

// ------------------------------- K1a: zero fill ----------------------------
__global__ __launch_bounds__(256) void zeroKernel(float4* __restrict__ p, int n4) {
  int i = blockIdx.x * 256 + threadIdx.x;
  if (i < n4) p[i] = make_float4(0.f, 0.f, 0.f, 0.f);
}

// ----------------------- K1b: per-row exclusive prefix ---------------------
__global__ __launch_bounds__(256) void rowPrefixKernel(const float* __restrict__ in,
                                                       float* __restrict__ pp) {
  __shared__ float sums[256];
  int row = blockIdx.x;              // 0..4095
  int b = row >> 10, y = row & 1023;
  const float4* rowIn = (const float4*)(in + ((size_t)b * W_IMG + y) * W_IMG);
  float* rowOut = pp + ((size_t)b * PP_ROWS + (y + 64)) * PP_PITCH;
  int t = threadIdx.x;

  float4 v = rowIn[t];
  float s0 = v.x, s1 = s0 + v.y, s2 = s1 + v.z, s3 = s2 + v.w;
  sums[t] = s3;
  __syncthreads();
  // Hillis-Steele inclusive scan of 256 chunk sums
  for (int off = 1; off < 256; off <<= 1) {
    float add = (t >= off) ? sums[t - off] : 0.f;
    __syncthreads();
    sums[t] += add;
    __syncthreads();
  }
  float excl = sums[t] - s3;
  float total = sums[255];
  rowOut[52 + 4*t + 0] = excl;
  rowOut[52 + 4*t + 1] = excl + s0;
  rowOut[52 + 4*t + 2] = excl + s1;
  rowOut[52 + 4*t + 3] = excl + s2;
  if (t == 0) rowOut[52 + 1024] = total;   // Pex[1024]
  if (t < 75) rowOut[1077 + t] = total;    // right clamp pad
}

// -------- K2: disk / annulus sums via prefix chords, TDM-staged LDS --------
// Block = 16x16 output tile. LDS tile: PP rows [y0-50, y0+66) x 120 cols
// starting at padded col (x0+2), fetched by the Tensor Data Mover.
__global__ __launch_bounds__(256) void diskKernel(const float* __restrict__ in,
                                                  const float* __restrict__ pp,
                                                  float* __restrict__ out,
                                                  float* __restrict__ me) {
  __shared__ float tile[116 * 120];   // 55,680 B
  int tileId = blockIdx.x;
  int b   = tileId >> 12;
  int rem = tileId & 4095;
  int y0 = (rem >> 6) << 4;
  int x0 = (rem & 63) << 4;

  const float* ppB = pp + (size_t)b * PP_ROWS * PP_PITCH;

  if (threadIdx.x < 32) {
    // Build Tensor DMA Descriptor (D#), ISA cdna5 ch.10.7 / 15.17.
    unsigned long long gaddr =
        (unsigned long long)(uintptr_t)(ppB + (size_t)(y0 + 14) * PP_PITCH + (x0 + 2));
    unsigned ldsOff = (unsigned)(uintptr_t)(&tile[0]);   // LDS aperture low 32b = byte offset
    u32x4 g0; i32x8 g1; i32x4 zz4 = {0, 0, 0, 0};
    i32x8 zz8 = {0, 0, 0, 0, 0, 0, 0, 0};
    g0[0] = 1u;                                          // count=1, user descriptor
    g0[1] = ldsOff;                                      // lds_addr
    g0[2] = (unsigned)(gaddr & 0xFFFFFFFFull);           // global_addr[31:0]
    g0[3] = (unsigned)((gaddr >> 32) & 0x1FFFFFFu) | (2u << 30);  // addr[56:32] | type=2
    g1[0] = (2 << 16);                 // data_size = 4B; no iterate/pad/multicast
    g1[1] = (PP_PITCH & 0xFFFF) << 16; // tensor_dim0[15:0]
    g1[2] = (PP_ROWS & 0xFFFF) << 16;  // tensor_dim0[31:16]=0 | tensor_dim1[15:0]
    g1[3] = (120 << 16);               // tensor_dim1[31:16]=0 | tile_dim0=120
    g1[4] = 116;                       // tile_dim1=116, tile_dim2=0
    g1[5] = PP_PITCH;                  // tensor_dim0_stride[31:0]
    g1[6] = 0;                         // stride hi / tensor_dim1_stride lo
    g1[7] = 0;
    __builtin_amdgcn_tensor_load_to_lds(g0, g1, zz4, zz4, zz8, 0);
    __builtin_amdgcn_s_wait_tensorcnt(0);
  }
  __syncthreads();

  int tx = threadIdx.x & 15, ty = threadIdx.x >> 4;
  int y = y0 + ty, x = x0 + tx;
  const int P = 120;

  // hillslope disk R=50 (101x101)
  float hsSum = 0.f, areaHS = 0.f;
  for (int dy = -50; dy <= 50; ++dy) {
    int w = (int)sqrtf((float)(2500 - dy * dy));
    const float* rowp = &tile[(ty + dy + 50) * P];
    hsSum  += rowp[tx + 51 + w] - rowp[tx + 50 - w];
    areaHS += (float)(2 * w + 1);
  }
  // smoothing disk R=11 (23x23)
  float smSum = 0.f, areaSM = 0.f;
  for (int dy = -11; dy <= 11; ++dy) {
    int w = (int)sqrtf((float)(121 - dy * dy));
    const float* rowp = &tile[(ty + dy + 50) * P];
    smSum  += rowp[tx + 51 + w] - rowp[tx + 50 - w];
    areaSM += (float)(2 * w + 1);
  }
  // annulus 2<=d<=5 = disk(5) minus 3x3 hole (points with d<2); area = 81-9 = 72
  float anSum = 0.f;
  for (int dy = -5; dy <= 5; ++dy) {
    int w = (int)sqrtf((float)(25 - dy * dy));
    const float* rowp = &tile[(ty + dy + 50) * P];
    anSum += rowp[tx + 51 + w] - rowp[tx + 50 - w];
  }
  for (int dy = -1; dy <= 1; ++dy) {
    const float* rowp = &tile[(ty + dy + 50) * P];
    anSum -= rowp[tx + 52] - rowp[tx + 49];   // w=1 chord
  }

  float in0 = in[((size_t)b * W_IMG + y) * W_IMG + x];
  float tpiHS = in0 - hsSum / (areaHS + 1e-12f);
  tpiHS = (fminf(fmaxf(tpiHS, -10.f), 10.f) + 10.f) * 0.05f;
  float tpiL = in0 - anSum / (72.0f + 1e-12f);
  tpiL = (fminf(fmaxf(tpiL, -10.f), 10.f) + 10.f) * 0.05f;
  float meV = smSum / (areaSM + 1e-12f);

  size_t o = (size_t)y * W_IMG + x;
  float* outB = out + (size_t)b * 6 * CH_STRIDE;
  outB[0 * CH_STRIDE + o] = tpiHS;
  outB[2 * CH_STRIDE + o] = tpiL;
  me[(size_t)b * CH_STRIDE + o] = meV;
}

// ------------- K3: separable 3x3 convs via V_WMMA_F32_16X16X4_F32 ----------
__device__ __forceinline__ float bcoef(int j, int n, float h0, float h1, float h2) {
  int d = j - n;
  return (d == 0) ? h0 : (d == 1) ? h1 : (d == 2) ? h2 : 0.0f;
}

// tile: 18x18 input (pitch 20) in per-wave LDS. T: 16x18 scratch (pitch 20).
// Vertical 3-tap in VALU, horizontal 3-tap as T(16xK) x BandedCoef(Kx16)
// accumulated over 5 K=4 chunks on the f32 matrix pipe.
__device__ __forceinline__ v8f sepConvWMMA(const float* tile, float* T, int lane,
                                           float v0, float v1, float v2,
                                           float h0, float h1, float h2) {
  for (int idx = lane; idx < 288; idx += 32) {       // 288 = 9*32, no divergence
    int m = idx / 18, c = idx - m * 18;
    T[m * 20 + c] = v0 * tile[m * 20 + c] + v1 * tile[(m + 1) * 20 + c]
                  + v2 * tile[(m + 2) * 20 + c];
  }
  int n  = lane & 15;
  int m  = lane & 15;
  int kk = (lane >> 4) << 1;                          // K lanes split: 0,1 | 2,3
  v8f acc = {0, 0, 0, 0, 0, 0, 0, 0};
#pragma unroll
  for (int c4 = 0; c4 < 5; ++c4) {
    int k0 = 4 * c4 + kk;
    v2f a, bf;
    a[0] = (k0     < 18) ? T[m * 20 + k0]     : 0.0f;
    a[1] = (k0 + 1 < 18) ? T[m * 20 + k0 + 1] : 0.0f;
    bf[0] = bcoef(k0,     n, h0, h1, h2);
    bf[1] = bcoef(k0 + 1, n, h0, h1, h2);
    acc = __builtin_amdgcn_wmma_f32_16x16x4_f32(false, a, false, bf,
                                                (short)0, acc, false, false);
  }
  return acc;
}

__global__ __launch_bounds__(256) void finalizeKernel(const float* __restrict__ in,
                                                      const float* __restrict__ me,
                                                      float* __restrict__ out) {
  __shared__ float lds[8 * 1040];     // per wave: dtm 360 | me 360 | T 320
  int lane = threadIdx.x & 31;
  int wv   = threadIdx.x >> 5;
  int tileId = blockIdx.x * 8 + wv;
  int b   = tileId >> 12;
  int rem = tileId & 4095;
  int y0 = (rem >> 6) << 4, x0 = (rem & 63) << 4;

  float* dtmT = &lds[wv * 1040];
  float* meT  = dtmT + 360;
  float* Tb   = meT + 360;
  const float* inB = in + (size_t)b * CH_STRIDE;
  const float* meB = me + (size_t)b * CH_STRIDE;

  for (int idx = lane; idx < 324; idx += 32) {        // 18x18 halo tiles, zero pad
    int r = idx / 18, c = idx - r * 18;
    int gy = y0 + r - 1, gx = x0 + c - 1;
    bool ok = (gy >= 0 && gy < 1024 && gx >= 0 && gx < 1024);
    size_t o = (size_t)gy * W_IMG + gx;
    dtmT[r * 20 + c] = ok ? inB[o] : 0.f;
    meT [r * 20 + c] = ok ? meB[o] : 0.f;
  }
  // per-wave LDS regions only; wave-internal DS ordering suffices (no barrier)

  // Sobel/8 on DTM; curvature kernels on mean elevation (all rank-1 separable)
  v8f dxA = sepConvWMMA(dtmT, Tb, lane,  1.f, 2.f, 1.f, -0.125f, 0.f,     0.125f);
  v8f dyA = sepConvWMMA(dtmT, Tb, lane, -1.f, 0.f, 1.f,  0.125f, 0.25f,   0.125f);
  v8f pA  = sepConvWMMA(meT,  Tb, lane,  1.f, 2.f, 1.f, -0.125f, 0.f,     0.125f);
  v8f qA  = sepConvWMMA(meT,  Tb, lane, -1.f, 0.f, 1.f,  0.125f, 0.25f,   0.125f);
  v8f rA  = sepConvWMMA(meT,  Tb, lane,  1.f, 1.f, 1.f,  1.f/3.f, -2.f/3.f, 1.f/3.f);
  v8f tA  = sepConvWMMA(meT,  Tb, lane,  1.f, -2.f, 1.f, 1.f/3.f,  1.f/3.f, 1.f/3.f);
  v8f sA  = sepConvWMMA(meT,  Tb, lane,  1.f, 0.f, -1.f, 0.25f,   0.f,    -0.25f);

  int n = lane & 15, hi = lane >> 4;
  float* outB = out + (size_t)b * 6 * CH_STRIDE;
#pragma unroll
  for (int g = 0; g < 8; ++g) {
    int mrow = g + hi * 8;                  // C/D layout: VGPR g -> M=g / M=g+8
    int y = y0 + mrow, x = x0 + n;
    size_t o = (size_t)y * W_IMG + x;
    float dx = dxA[g], dy = dyA[g];
    float g2 = dx * dx + dy * dy;
    float slpR = atanf(sqrtf(g2));
    float slp  = fminf(sqrtf(slpR * 57.2958f), 10.f) * 0.1f;
    // sum of cos(AZ_i - aspect) over the 4 azimuths is identically zero, so
    // hillshade = cos(45deg)*cos(slpR) = 0.7071 * rsqrt(1 + |grad|^2)
    float hill = 0.70710678f * rsqrtf(1.f + g2);
    float p = pA[g], q = qA[g], r = rA[g], t = tA[g], s = sA[g];
    float pq2 = p * p + q * q;
    float denom = pq2 * sqrtf(pq2) + 1e-12f;
    float crvPln = (q * q * r - 2.f * p * q * s + p * p * t) / denom;
    float crvPro = (p * p * r + 2.f * p * q * s + q * p * t) / denom;
    crvPln = (fminf(fmaxf(crvPln, -0.1f), 0.1f) + 0.1f) * 5.f;
    crvPro = (fminf(fmaxf(crvPro, -0.1f), 0.1f) + 0.1f) * 5.f;
    outB[1 * CH_STRIDE + o] = slp;
    outB[3 * CH_STRIDE + o] = hill;
    outB[4 * CH_STRIDE + o] = crvPro;
    outB[5 * CH_STRIDE + o] = crvPln;
  }
}

// ---------------------------------------------------------------------------
extern "C" void kernel_launch(void* const* d_in, const int* in_sizes, int n_in,
                              void* d_out, int out_size, void* d_ws, size_t ws_size,
                              hipStream_t stream) {
  (void)in_sizes; (void)n_in; (void)out_size; (void)ws_size;
  const float* in = (const float*)d_in[0];
  float* out = (float*)d_out;
  float* pp = (float*)d_ws;                           // 4*1154*1152*4B = 21.3 MB
  size_t ppElems = (size_t)4 * PP_ROWS * PP_PITCH;
  float* me = pp + ppElems;                           // 4*1024*1024*4B = 16.8 MB

  int n4 = (int)(ppElems / 4);
  zeroKernel<<<(n4 + 255) / 256, 256, 0, stream>>>((float4*)pp, n4);
  rowPrefixKernel<<<4 * 1024, 256, 0, stream>>>(in, pp);
  diskKernel<<<4 * 64 * 64, 256, 0, stream>>>(in, pp, out, me);
  finalizeKernel<<<(4 * 64 * 64) / 8, 256, 0, stream>>>(in, me, out);
}